// CrossModalFusion_2018634629679
// MI455X (gfx1250) — compile-verified
//
#include <hip/hip_runtime.h>
#include <math.h>

// ---------------------------------------------------------------------------
// CrossModalFusion for MI455X (gfx1250, wave32).
// Memory-bound problem: H is 134 MB (fits in 192 MB L2); everything else ~5MB.
// Algebra: score_n = H_n . (Wk^T Q_b)  (bk cancels in softmax),
//          H~_b    = Wv @ (sum attn_n H_n) + bv  (since sum attn = 1).
// => 2 streaming passes over H (2nd pass hits L2) + small fp32 WMMA GEMMs.
// ---------------------------------------------------------------------------

typedef __attribute__((ext_vector_type(2))) float v2f;
typedef __attribute__((ext_vector_type(8))) float v8f;

#define DV 256

// ---- fp32 WMMA GEMM: D[M,N] = alpha * (A[M,K] x B) + bias ------------------
// BT=true : B stored as [N,K] row-major (i.e. weight W, out = A @ W^T)
// BT=false: B stored as [K,N] row-major
// One wave per 16x16 output tile; K stepped by 4 via v_wmma_f32_16x16x4_f32.
template <bool BT>
__global__ void __launch_bounds__(128)
cmf_wmma_gemm_f32(const float* __restrict__ A, const float* __restrict__ Bm,
                  const float* __restrict__ bias, float* __restrict__ D,
                  int N, int K, int lda, int ldb, int ldd, float alpha) {
  const int wave = threadIdx.x >> 5;
  const int lane = threadIdx.x & 31;
  const int tile = blockIdx.x * (blockDim.x >> 5) + wave;
  const int ntn  = N >> 4;
  const int tm   = tile / ntn;
  const int tn   = tile - tm * ntn;
  const int r0   = tm << 4, n0 = tn << 4;
  const int lr   = lane & 15;      // row (A) / col (B) within tile
  const int hi   = lane >> 4;      // selects K pair {0,1} vs {2,3}

  v8f acc = {};
  const float* ap = A + (size_t)(r0 + lr) * lda + (hi << 1);
  const float* bp = BT ? (Bm + (size_t)(n0 + lr) * ldb + (hi << 1)) : Bm;

#pragma unroll 4
  for (int k0 = 0; k0 < K; k0 += 4) {
    v2f a = *(const v2f*)(ap + k0);
    v2f b;
    if (BT) {
      b = *(const v2f*)(bp + k0);
    } else {
      const int kk = k0 + (hi << 1);
      b.x = Bm[(size_t)kk * ldb + n0 + lr];
      b.y = Bm[(size_t)(kk + 1) * ldb + n0 + lr];
    }
    acc = __builtin_amdgcn_wmma_f32_16x16x4_f32(
        false, a, false, b, (short)0, acc, false, false);
  }

  const int col = n0 + lr;
  const float bc = bias ? bias[col] : 0.0f;
#pragma unroll
  for (int r = 0; r < 8; ++r) {
    const int row = r0 + r + (hi << 3);  // C/D layout: vgpr r -> M=r (+8 for hi lanes)
    D[(size_t)row * ldd + col] = acc[r] * alpha + bc;
  }
}

// ---- segment offsets via binary search on sorted batch_idx ----------------
__global__ void cmf_offsets(const int* __restrict__ bidx, int* __restrict__ offs,
                            int N, int B) {
  const int b = blockIdx.x * blockDim.x + threadIdx.x;
  if (b > B) return;
  int lo = 0, hi = N;
  while (lo < hi) {
    const int mid = (lo + hi) >> 1;
    if (bidx[mid] < b) lo = mid + 1; else hi = mid;
  }
  offs[b] = lo;  // offs[B] = N (all values < B)
}

// ---- pass 1 over H: scores[n] = H_n . qt[batch_idx[n]]  (qt pre-scaled) ---
// one wave per node; lane l covers 8 contiguous floats -> wave reads 1KB row
__global__ void cmf_scores(const float* __restrict__ H, const float* __restrict__ qt,
                           const int* __restrict__ bidx, float* __restrict__ scores,
                           int N) {
  const int wid  = (blockIdx.x * blockDim.x + threadIdx.x) >> 5;
  const int lane = threadIdx.x & 31;
  const int nw   = (gridDim.x * blockDim.x) >> 5;
  for (int n = wid; n < N; n += nw) {
    const float* h = H + (size_t)n * DV + lane * 8;
    const float* q = qt + (size_t)bidx[n] * DV + lane * 8;
    const float4 h0 = *(const float4*)h;
    const float4 h1 = *(const float4*)(h + 4);
    const float4 q0 = *(const float4*)q;
    const float4 q1 = *(const float4*)(q + 4);
    float acc = h0.x * q0.x + h0.y * q0.y + h0.z * q0.z + h0.w * q0.w +
                h1.x * q1.x + h1.y * q1.y + h1.z * q1.z + h1.w * q1.w;
#pragma unroll
    for (int off = 16; off > 0; off >>= 1) acc += __shfl_xor(acc, off, 32);
    if (lane == 0) scores[n] = acc;
  }
}

// ---- segment softmax: attn[n] = exp(s-m)/denom per contiguous segment -----
__global__ void __launch_bounds__(256)
cmf_softmax(const float* __restrict__ scores, const int* __restrict__ offs,
            float* __restrict__ attn) {
  const int b = blockIdx.x;
  const int s = offs[b], e = offs[b + 1];
  __shared__ float red[256];
  const int t = threadIdx.x;

  float m = -INFINITY;
  for (int n = s + t; n < e; n += 256) m = fmaxf(m, scores[n]);
  red[t] = m; __syncthreads();
  for (int w = 128; w > 0; w >>= 1) {
    if (t < w) red[t] = fmaxf(red[t], red[t + w]);
    __syncthreads();
  }
  m = red[0]; __syncthreads();

  float d = 0.0f;
  for (int n = s + t; n < e; n += 256) d += __expf(scores[n] - m);
  red[t] = d; __syncthreads();
  for (int w = 128; w > 0; w >>= 1) {
    if (t < w) red[t] += red[t + w];
    __syncthreads();
  }
  d = red[0];
  if (e <= s) return;  // empty graph: nothing to write
  const float inv = 1.0f / d;
  for (int n = s + t; n < e; n += 256) attn[n] = __expf(scores[n] - m) * inv;
}

// ---- pass 2 over H (L2-resident): S_b = sum attn_n * H_n ------------------
// one block per graph; thread t owns column t -> fully coalesced row reads
__global__ void __launch_bounds__(256)
cmf_wsum(const float* __restrict__ H, const float* __restrict__ attn,
         const int* __restrict__ offs, float* __restrict__ S) {
  const int b = blockIdx.x, t = threadIdx.x;
  const int s = offs[b], e = offs[b + 1];
  float acc = 0.0f;
  int n = s;
  for (; n + 4 <= e; n += 4) {
    acc += attn[n]     * H[(size_t)n * DV + t];
    acc += attn[n + 1] * H[(size_t)(n + 1) * DV + t];
    acc += attn[n + 2] * H[(size_t)(n + 2) * DV + t];
    acc += attn[n + 3] * H[(size_t)(n + 3) * DV + t];
  }
  for (; n < e; ++n) acc += attn[n] * H[(size_t)n * DV + t];
  S[(size_t)b * DV + t] = acc;  // empty segment -> 0, fixed up later
}

// ---- U[:, :256] = V -------------------------------------------------------
__global__ void cmf_build_u(const float* __restrict__ V, float* __restrict__ U,
                            int total) {
  const int i = blockIdx.x * blockDim.x + threadIdx.x;
  if (i >= total) return;
  const int b = i >> 8, d = i & 255;
  U[(size_t)b * (2 * DV) + d] = V[i];
}

// ---- empty graphs pass V through unchanged --------------------------------
__global__ void cmf_fixup(const float* __restrict__ V, const int* __restrict__ offs,
                          float* __restrict__ out, int total) {
  const int i = blockIdx.x * blockDim.x + threadIdx.x;
  if (i >= total) return;
  const int b = i >> 8;
  if (offs[b + 1] == offs[b]) out[i] = V[i];
}

extern "C" void kernel_launch(void* const* d_in, const int* in_sizes, int n_in,
                              void* d_out, int out_size, void* d_ws, size_t ws_size,
                              hipStream_t stream) {
  (void)n_in; (void)out_size; (void)ws_size;
  const float* V    = (const float*)d_in[0];
  const float* H    = (const float*)d_in[1];
  const int*   bidx = (const int*)d_in[2];
  const float* Wq   = (const float*)d_in[3];
  const float* bq   = (const float*)d_in[4];
  const float* Wk   = (const float*)d_in[5];
  // d_in[6] = bk: cancels under softmax shift-invariance, unused
  const float* Wv   = (const float*)d_in[7];
  const float* bv   = (const float*)d_in[8];
  const float* Wo   = (const float*)d_in[9];
  const float* bo   = (const float*)d_in[10];
  float* out = (float*)d_out;

  const int B = in_sizes[0] / DV;   // 1024
  const int N = in_sizes[1] / DV;   // 131072
  const float scale = 1.0f / 16.0f; // 1/sqrt(256)

  // workspace layout (floats)
  float* Q      = (float*)d_ws;                    // B*256
  float* qt     = Q + (size_t)B * DV;              // B*256
  float* scores = qt + (size_t)B * DV;             // N
  float* attn   = scores + N;                      // N
  float* S      = attn + N;                        // B*256
  float* U      = S + (size_t)B * DV;              // B*512
  int*   offs   = (int*)(U + (size_t)B * 2 * DV);  // B+1

  const int tilesB  = (B / 16) * (DV / 16);        // tiles for [B,256] outputs
  const int gemmBlk = tilesB / 4;                  // 4 waves (16x16 tiles) / block

  // 1) Q = V @ Wq^T + bq
  cmf_wmma_gemm_f32<true><<<gemmBlk, 128, 0, stream>>>(
      V, Wq, bq, Q, DV, DV, DV, DV, DV, 1.0f);
  // 2) qt = (Q @ Wk) * scale
  cmf_wmma_gemm_f32<false><<<gemmBlk, 128, 0, stream>>>(
      Q, Wk, (const float*)nullptr, qt, DV, DV, DV, DV, DV, scale);
  // 3) segment offsets (batch_idx sorted)
  cmf_offsets<<<(B + 1 + 255) / 256, 256, 0, stream>>>(bidx, offs, N, B);
  // 4) pass 1 over H: scores (HBM-bound, fills L2)
  cmf_scores<<<(N * 32) / 256, 256, 0, stream>>>(H, qt, bidx, scores, N);
  // 5) segment softmax -> attn
  cmf_softmax<<<B, 256, 0, stream>>>(scores, offs, attn);
  // 6) pass 2 over H (L2 resident): S_b = sum attn_n H_n
  cmf_wsum<<<B, 256, 0, stream>>>(H, attn, offs, S);
  // 7) U = [V | H~],  H~ = S @ Wv^T + bv
  cmf_build_u<<<(B * DV + 255) / 256, 256, 0, stream>>>(V, U, B * DV);
  cmf_wmma_gemm_f32<true><<<gemmBlk, 128, 0, stream>>>(
      S, Wv, bv, U + DV, DV, DV, DV, DV, 2 * DV, 1.0f);
  // 8) out = U @ Wo^T + bo   (K = 512)
  cmf_wmma_gemm_f32<true><<<gemmBlk, 128, 0, stream>>>(
      U, Wo, bo, out, DV, 2 * DV, 2 * DV, 2 * DV, DV, 1.0f);
  // 9) graphs with zero nodes pass V through
  cmf_fixup<<<(B * DV + 255) / 256, 256, 0, stream>>>(V, offs, out, B * DV);
}